// HungarianMatcher_66486093742637
// MI455X (gfx1250) — compile-verified
//
#include <hip/hip_runtime.h>
#include <hip/hip_bf16.h>

typedef __attribute__((ext_vector_type(2))) float v2f;
typedef __attribute__((ext_vector_type(8))) float v8f;

#define ALPHA_F 0.25f
#define EPS_F   1e-8f

// ---------------------------------------------------------------------------
// Kernel 1: per-(row,class) focal classification diff = pos_cost - neg_cost.
// 230K elements -> keeps all transcendentals out of the 55M-entry main loop.
// ---------------------------------------------------------------------------
__global__ void hm_prep_classdiff(const float* __restrict__ logits,
                                  float* __restrict__ diff, int n) {
  int i = blockIdx.x * blockDim.x + threadIdx.x;
  if (i < n) {
    float x = logits[i];
    float p = 1.0f / (1.0f + __expf(-x));
    float neg = (1.0f - ALPHA_F) * p * p * (-__logf(1.0f - p + EPS_F));
    float pos = ALPHA_F * (1.0f - p) * (1.0f - p) * (-__logf(p + EPS_F));
    diff[i] = pos - neg;
  }
}

// ---------------------------------------------------------------------------
// Kernel 2: cost matrix. Block = 8 waves, covers 16 rows x 128 targets.
// Each wave computes one 16x16 tile:
//   cost_class tile = Diff[16x8] @ OneHot[8x16]  via 2x V_WMMA_F32_16X16X4_F32
// then adds L1 + (-GIoU) per entry into the f32 accumulator layout and stores.
// ---------------------------------------------------------------------------
__global__ __launch_bounds__(256) void hm_cost_kernel(
    const float* __restrict__ cdiff,   // [M,8] focal class diffs
    const float* __restrict__ pboxes,  // [M,4] cxcywh
    const int*   __restrict__ tids,    // [T]
    const float* __restrict__ tboxes,  // [T,4] cxcywh
    float* __restrict__ out, int M, int T)
{
  const int lane = threadIdx.x & 31;
  const int wave = threadIdx.x >> 5;
  const int half = lane >> 4;                       // 0: lanes 0-15, 1: 16-31
  const int m0   = blockIdx.y << 4;
  const int t    = (blockIdx.x << 7) + (wave << 4) + (lane & 15);

  // Stage the block's 16 pred boxes in LDS (shared by all 8 waves).
  __shared__ float4 s_pbox[16];
  if (threadIdx.x < 16) {
    int m = m0 + threadIdx.x;
    if (m >= M) m = M - 1;
    s_pbox[threadIdx.x] = reinterpret_cast<const float4*>(pboxes)[m];
  }
  __syncthreads();

  // ---- A fragment: 16x4 f32 layout (lanes 0-15: K={0,1}; 16-31: K={2,3})
  int am = m0 + (lane & 15);
  if (am >= M) am = M - 1;
  const float* dp = cdiff + (size_t)am * 8 + (half ? 2 : 0);
  v2f a_lo, a_hi;
  a_lo.x = dp[0]; a_lo.y = dp[1];   // K = {0,1} / {2,3}
  a_hi.x = dp[4]; a_hi.y = dp[5];   // K = {4,5} / {6,7}

  // ---- B fragment: 4x16 one-hot of target class (row-striped: VGPR0 holds
  //      K=0 in lanes 0-15 / K=2 in lanes 16-31; VGPR1 holds K=1 / K=3)
  int tid = (t < T) ? tids[t] : -1;
  int kb  = half ? 2 : 0;
  v2f b_lo, b_hi;
  b_lo.x = (tid == kb + 0) ? 1.0f : 0.0f;
  b_lo.y = (tid == kb + 1) ? 1.0f : 0.0f;
  b_hi.x = (tid == kb + 4) ? 1.0f : 0.0f;
  b_hi.y = (tid == kb + 5) ? 1.0f : 0.0f;

  v8f acc = {};
  acc = __builtin_amdgcn_wmma_f32_16x16x4_f32(false, a_lo, false, b_lo,
                                              (short)0, acc, false, false);
  acc = __builtin_amdgcn_wmma_f32_16x16x4_f32(false, a_hi, false, b_hi,
                                              (short)0, acc, false, false);

  // ---- per-column target box derived values (one box per lane)
  int tc = (t < T) ? t : 0;
  float4 tb = reinterpret_cast<const float4*>(tboxes)[tc];
  float tx1 = tb.x - 0.5f * tb.z, ty1 = tb.y - 0.5f * tb.w;
  float tx2 = tb.x + 0.5f * tb.z, ty2 = tb.y + 0.5f * tb.w;
  float tarea = (tx2 - tx1) * (ty2 - ty1);

  // ---- epilogue: 8 (row, col) entries per lane matching the D-matrix layout
  //      (VGPR r -> row r for lanes 0-15, row r+8 for lanes 16-31)
  #pragma unroll
  for (int r = 0; r < 8; ++r) {
    float4 pb = s_pbox[(half << 3) + r];
    float l1 = fabsf(pb.x - tb.x) + fabsf(pb.y - tb.y) +
               fabsf(pb.z - tb.z) + fabsf(pb.w - tb.w);
    float px1 = pb.x - 0.5f * pb.z, py1 = pb.y - 0.5f * pb.w;
    float px2 = pb.x + 0.5f * pb.z, py2 = pb.y + 0.5f * pb.w;
    float parea = (px2 - px1) * (py2 - py1);
    float iw = fmaxf(fminf(px2, tx2) - fmaxf(px1, tx1), 0.0f);
    float ih = fmaxf(fminf(py2, ty2) - fmaxf(py1, ty1), 0.0f);
    float inter = iw * ih;
    float uni   = parea + tarea - inter;
    float iou   = inter / (uni + EPS_F);
    float ew = fmaxf(fmaxf(px2, tx2) - fminf(px1, tx1), 0.0f);
    float eh = fmaxf(fmaxf(py2, ty2) - fminf(py1, ty1), 0.0f);
    float earea = ew * eh;
    float giou  = iou - (earea - uni) / (earea + EPS_F);
    acc[r] += l1 - giou;   // W_BBOX*l1 + W_GIOU*(-giou), weights == 1
  }

  if (t < T) {
    #pragma unroll
    for (int r = 0; r < 8; ++r) {
      int m = m0 + (half << 3) + r;
      if (m < M) out[(size_t)m * (size_t)T + t] = acc[r];
    }
  }
}

extern "C" void kernel_launch(void* const* d_in, const int* in_sizes, int n_in,
                              void* d_out, int out_size, void* d_ws, size_t ws_size,
                              hipStream_t stream) {
  const float* logits = (const float*)d_in[0];   // [B,N,C]
  const float* pboxes = (const float*)d_in[1];   // [B,N,4]
  const int*   tids   = (const int*)d_in[2];     // [T]
  const float* tboxes = (const float*)d_in[3];   // [T,4]
  float* out = (float*)d_out;

  int MC = in_sizes[0];        // B*N*C  (C == 8)
  int M  = in_sizes[1] / 4;    // B*N
  int T  = in_sizes[2];

  float* cdiff = (float*)d_ws; // M*8 floats (~921 KB)

  hm_prep_classdiff<<<(MC + 255) / 256, 256, 0, stream>>>(logits, cdiff, MC);

  dim3 grid((T + 127) / 128, (M + 15) / 16);
  hm_cost_kernel<<<grid, 256, 0, stream>>>(cdiff, pboxes, tids, tboxes, out, M, T);
}